// MetaSelectTarget_61409442398236
// MI455X (gfx1250) — compile-verified
//
#include <hip/hip_runtime.h>

// MetaSelectTarget: FCOS-style level assignment.
// One block per (b, m) GT box: 160 threads = 5 wave32s, wave l handles level l.
// Lanes 0..19 each own one float4 chunk of the 80-class row (coalesced 320B),
// lane 20 owns the IoU/regression term. Per-lane accumulation across the
// shrunk-region cells, then a single V_WMMA_F32_16X16X4_F32 vs. all-ones B
// performs the 32-lane reduction (D[m,n] = v_m + v_{m+16}); summing the 8 D
// VGPRs + one xor-16 shuffle yields the full wave sum in every lane.

#define TOTAL_LOCS 21824   // 128^2+64^2+32^2+16^2+8^2
#define NLEV 5
#define M_BOXES 32
#define BIGLOSS 1.0e7f
#define EPS_F 1.0e-6f

typedef __attribute__((ext_vector_type(2))) float v2f;
typedef __attribute__((ext_vector_type(8))) float v8f;

__device__ __forceinline__ float wave_reduce_sum_wmma(float v) {
  // A: both K-slots of this lane hold v/2 -> row m of A sums to v_m + v_{m+16}
  v2f a; a.x = v * 0.5f; a.y = v * 0.5f;
  v2f b; b.x = 1.0f;     b.y = 1.0f;      // B = all ones (layout-independent)
  v8f c = {};
  v8f d = __builtin_amdgcn_wmma_f32_16x16x4_f32(
      /*neg_a=*/false, a, /*neg_b=*/false, b,
      /*c_mod=*/(short)0, c, /*reuse_a=*/false, /*reuse_b=*/false);
  // lane n<16 regs hold D[0..7, n], lane n>=16 hold D[8..15, n-16]; D indep of n
  float s = d[0] + d[1] + d[2] + d[3] + d[4] + d[5] + d[6] + d[7];
  s += __shfl_xor(s, 16, 32);  // combine the two row-halves -> full 32-lane sum
  return s;
}

__global__ __launch_bounds__(160)
void meta_select_target_kernel(const float* __restrict__ cls,
                               const float* __restrict__ regr,
                               const float* __restrict__ gt,
                               int* __restrict__ out) {
  const int bm    = blockIdx.x;            // (b, m) pair, 0..511
  const int b     = bm / M_BOXES;
  const int lane  = threadIdx.x & 31;
  const int level = threadIdx.x >> 5;      // 0..4, one wave per level

  // --- GT box (uniform within block) ---
  const float* g = gt + (size_t)bm * 5;
  const float x1 = g[0], y1 = g[1], x2 = g[2], y2 = g[3], lf = g[4];
  const bool valid = (x1 != 0.f) || (y1 != 0.f) || (x2 != 0.f) ||
                     (y2 != 0.f) || (lf != 0.f);
  const int label = (int)lf;

  // --- level geometry (uniform within wave) ---
  const int stride = 8 << level;           // 8,16,32,64,128
  const int fw     = 128 >> level;         // square level: fh == fw
  int start = 0;
  #pragma unroll
  for (int j = 0; j < NLEV; ++j) {
    if (j < level) { const int f = 128 >> j; start += f * f; }
  }

  const float inv_s = 1.0f / (float)stride;          // exact (power of 2)
  const float bx1 = x1 * inv_s, by1 = y1 * inv_s;
  const float bx2 = x2 * inv_s, by2 = y2 * inv_s;
  const float cxs = (bx1 + bx2) * 0.5f, cys = (by1 + by2) * 0.5f;
  const float hw  = (bx2 - bx1) * 0.2f,  hh  = (by2 - by1) * 0.2f;
  const float fwf = (float)fw;
  const int px1 = (int)fminf(fmaxf(floorf(cxs - hw * 0.5f), 0.f), fwf);
  const int px2 = (int)fminf(fmaxf(ceilf (cxs + hw * 0.5f), 0.f), fwf);
  const int py1 = (int)fminf(fmaxf(floorf(cys - hh * 0.5f), 0.f), fwf);
  const int py2 = (int)fminf(fmaxf(ceilf (cys + hh * 0.5f), 0.f), fwf);

  const bool degen  = (px1 == px2) || (py1 == py2);
  const bool active = valid && !degen;     // wave-uniform branch

  float acc = 0.f;
  float cnt = 1.f;
  if (active) {
    const int nx = px2 - px1, ny = py2 - py1;
    const int ncells = nx * ny;
    cnt = (float)ncells;
    const size_t base = (size_t)b * TOTAL_LOCS;

    for (int i = 0; i < ncells; ++i) {
      const int cy  = py1 + i / nx;
      const int cx  = px1 + i % nx;
      const int loc = start + cy * fw + cx;
      const float* row = cls + (base + (size_t)loc) * 80;

      if (i + 1 < ncells) {  // global_prefetch_b8 for next cell's class row
        const int i2 = i + 1;
        const int loc2 = start + (py1 + i2 / nx) * fw + (px1 + i2 % nx);
        __builtin_prefetch(cls + (base + (size_t)loc2) * 80, 0, 3);
      }

      if (lane < 20) {
        // focal classification: neg_sum over this lane's 4 classes,
        // plus (pos - neg) at the label class if it lands in this chunk.
        const float4 q = *(const float4*)(row + lane * 4);
        const float xs[4] = {q.x, q.y, q.z, q.w};
        #pragma unroll
        for (int j = 0; j < 4; ++j) {
          const float x   = xs[j];
          const float t   = expf(-fabsf(x));
          const float l1p = log1pf(t);
          const float inv = 1.0f / (1.0f + t);
          const float p   = (x >= 0.f) ? inv : t * inv;
          const float log1mp = fminf(-x, 0.f) - l1p;    // log_sigmoid(-x)
          const float negv = -0.75f * p * p * log1mp;   // (1-ALPHA)=0.75
          acc += negv;
          if (lane * 4 + j == label) {
            const float logp = fminf(x, 0.f) - l1p;     // log_sigmoid(x)
            const float omp  = 1.0f - p;
            const float posv = -0.25f * omp * omp * logp;
            acc += posv - negv;
          }
        }
      } else if (lane == 20) {
        // IoU regression loss at this cell
        const float4 r = *(const float4*)(regr + (base + (size_t)loc) * 4);
        const float sx = ((float)cx + 0.5f) * (float)stride;
        const float sy = ((float)cy + 0.5f) * (float)stride;
        const float tl = (sx - x1) * 0.25f;
        const float tt = (sy - y1) * 0.25f;
        const float tr = (x2 - sx) * 0.25f;
        const float tb = (y2 - sy) * 0.25f;
        const float area_p = (r.x + r.z) * (r.y + r.w);
        const float area_t = (tl + tr) * (tt + tb);
        const float iw = fmaxf(fminf(r.x, tl) + fminf(r.z, tr), 0.f);
        const float ih = fmaxf(fminf(r.y, tt) + fminf(r.w, tb), 0.f);
        const float inter = iw * ih;
        const float uni   = area_p + area_t - inter;
        float iou = inter / fmaxf(uni, EPS_F);
        iou = fminf(fmaxf(iou, EPS_F), 1.0f);
        acc += -logf(iou);
      }
    }
  }

  // EXEC is all-ones here (uniform branches reconverged) -> WMMA legal.
  const float total = wave_reduce_sum_wmma(acc);
  const float loss  = active ? (total / cnt) : BIGLOSS;

  __shared__ float lloss[NLEV];
  if (lane == 0) lloss[level] = loss;
  __syncthreads();

  if (threadIdx.x == 0) {
    int best = 0;
    float bv = lloss[0];
    #pragma unroll
    for (int l = 1; l < NLEV; ++l) {
      if (lloss[l] < bv) { bv = lloss[l]; best = l; }   // first-min semantics
    }
    out[bm] = valid ? best : -1;
  }
}

extern "C" void kernel_launch(void* const* d_in, const int* in_sizes, int n_in,
                              void* d_out, int out_size, void* d_ws, size_t ws_size,
                              hipStream_t stream) {
  // inputs (setup_inputs order): [0] dummy scalar, [1] batch_cls_pred f32,
  // [2] batch_regr_pred f32, [3] feature_shapes i32 (unused; constants match),
  // [4] batch_gt_boxes f32
  const float* cls  = (const float*)d_in[1];
  const float* regr = (const float*)d_in[2];
  const float* gt   = (const float*)d_in[4];
  int* out = (int*)d_out;

  const int n_bm = out_size;               // B*M = 512 tasks
  dim3 grid(n_bm), block(160);             // 5 wave32s per block (one per level)
  hipLaunchKernelGGL(meta_select_target_kernel, grid, block, 0, stream,
                     cls, regr, gt, out);
}